// HighOrder_ACM_Framework_52012053954565
// MI455X (gfx1250) — compile-verified
//
#include <hip/hip_runtime.h>
#include <hip/hip_bf16.h>
#include <math.h>

#define N_NODES 50000
#define N_EDGES 800000
#define DIM 64

typedef __attribute__((ext_vector_type(2))) float v2f;
typedef __attribute__((ext_vector_type(8))) float v8f;

// ---------------------------------------------------------------------------
// Graph phase kernels (L2-resident scatter/gather; memory/atomic bound)
// ---------------------------------------------------------------------------

__global__ void k_deg(const int* __restrict__ ei, float* __restrict__ deg) {
  int e = blockIdx.x * blockDim.x + threadIdx.x;
  if (e < N_EDGES) {
    int d = ei[N_EDGES + e];  // edge_index[1][e] = dst
    atomicAdd(&deg[d], 1.0f);
  }
}

__global__ void k_inv_deg(const float* __restrict__ deg, float* __restrict__ inv_deg) {
  int i = blockIdx.x * blockDim.x + threadIdx.x;
  if (i < N_NODES) inv_deg[i] = 1.0f / fmaxf(deg[i], 1.0f);
}

// one thread per (edge, feature): lanes of a wave cover 32 consecutive floats of
// one node row -> coalesced gather, row-contiguous float atomics to L2
__global__ void k_scatter(const float* __restrict__ feat, const int* __restrict__ ei,
                          float* __restrict__ agg) {
  int t = blockIdx.x * blockDim.x + threadIdx.x;  // E*64 = 51.2M < 2^31
  if (t < N_EDGES * DIM) {
    int e = t >> 6;
    int f = t & 63;
    int s = ei[e];             // src
    int d = ei[N_EDGES + e];   // dst
    atomicAdd(&agg[(size_t)d * DIM + f], feat[(size_t)s * DIM + f]);
  }
}

__global__ void k_scale(float* __restrict__ agg, const float* __restrict__ inv_deg) {
  int t = blockIdx.x * blockDim.x + threadIdx.x;
  if (t < N_NODES * DIM) agg[t] *= inv_deg[t >> 6];
}

// ---------------------------------------------------------------------------
// Fused dense phase: 5x (GEMM 64x64 + bias + ReLU + gate + scaled accumulate)
// One wave32 per 16-row node block, V_WMMA_F32_16X16X4_F32 on the matrix pipe.
// ---------------------------------------------------------------------------

struct BranchParams {
  const float* W[5];   // slots in setup order: hp, lp, i, hp2, lp2
  const float* b[5];
  const float* wa[5];
  const float* ba[5];
};

__launch_bounds__(32)
__global__ void k_acm_wmma(const float* __restrict__ x, const float* __restrict__ ax,
                           const float* __restrict__ aax, BranchParams p,
                           float* __restrict__ out) {
  // Reference branch order: hp, lp, hp2, lp2, i   (param slot via PIDX)
  const int   PIDX[5] = {0, 1, 3, 4, 2};
  const float CX[5]   = {1.f, 0.f, 1.f, 0.f, 1.f};   // coeff on x
  const float CA[5]   = {-1.f, 1.f, -2.f, 0.f, 0.f}; // coeff on ax
  const float CAA[5]  = {0.f, 0.f, 1.f, 1.f, 0.f};   // coeff on aax

  __shared__ float sH[16][65];   // padded: conflict-free row reads in gate dot
  __shared__ float sAlpha[16];

  const int lane = threadIdx.x;
  const int c = lane & 15;   // col within 16-wide tile / row for A-frag
  const int h = lane >> 4;   // lane half
  const int r0 = blockIdx.x * 16;

  const float* xr   = x   + (size_t)(r0 + c) * DIM;
  const float* axr  = ax  + (size_t)(r0 + c) * DIM;
  const float* aaxr = aax + (size_t)(r0 + c) * DIM;

  v8f oacc[4];
#pragma unroll
  for (int ct = 0; ct < 4; ++ct)
    oacc[ct] = (v8f){0.f, 0.f, 0.f, 0.f, 0.f, 0.f, 0.f, 0.f};

#pragma unroll
  for (int br = 0; br < 5; ++br) {
    const int pi = PIDX[br];
    const float cx = CX[br], ca = CA[br], caa = CAA[br];
    const float* __restrict__ W    = p.W[pi];   // [64][64] row-major (k, n)
    const float* __restrict__ bias = p.b[pi];   // [64]
    const float* __restrict__ wa   = p.wa[pi];  // [64]
    const float  ba = p.ba[pi][0];

    v8f acc[4];
#pragma unroll
    for (int ct = 0; ct < 4; ++ct)
      acc[ct] = (v8f){0.f, 0.f, 0.f, 0.f, 0.f, 0.f, 0.f, 0.f};

    // K = 64 in 16 steps of 4; A-frag built on the fly (constants fold per branch)
#pragma unroll
    for (int k = 0; k < 16; ++k) {
      const int kb = k * 4 + h * 2;  // this lane's K pair within the step
      v2f a;
      a.x = cx * xr[kb]     + ca * axr[kb]     + caa * aaxr[kb];
      a.y = cx * xr[kb + 1] + ca * axr[kb + 1] + caa * aaxr[kb + 1];
#pragma unroll
      for (int ct = 0; ct < 4; ++ct) {
        v2f bf;
        bf.x = W[(size_t)kb * DIM       + ct * 16 + c];
        bf.y = W[(size_t)(kb + 1) * DIM + ct * 16 + c];
        acc[ct] = __builtin_amdgcn_wmma_f32_16x16x4_f32(
            false, a, false, bf, (short)0, acc[ct], false, false);
      }
    }

    // bias + ReLU; stash H tile to LDS for the gating dot
#pragma unroll
    for (int ct = 0; ct < 4; ++ct) {
#pragma unroll
      for (int v = 0; v < 8; ++v) {
        float hv = acc[ct][v] + bias[ct * 16 + c];
        hv = fmaxf(hv, 0.f);
        acc[ct][v] = hv;                       // keep for scaled accumulate
        sH[v + 8 * h][ct * 16 + c] = hv;       // row = v + 8h, col = ct*16+c
      }
    }
    __syncthreads();

    // gate: alpha[row] = sigmoid(dot(H[row,:], wa) + ba)
    float partial = 0.f;
#pragma unroll
    for (int j = 0; j < 32; ++j) {
      int f = h * 32 + j;
      partial += sH[c][f] * wa[f];  // lane's row = c; halves split the 64 feats
    }
    partial += __shfl_xor(partial, 16, 32);
    if (h == 0) sAlpha[c] = 1.0f / (1.0f + __expf(-(partial + ba)));
    __syncthreads();

#pragma unroll
    for (int ct = 0; ct < 4; ++ct)
#pragma unroll
      for (int v = 0; v < 8; ++v)
        oacc[ct][v] += sAlpha[v + 8 * h] * acc[ct][v];
    __syncthreads();  // sH reused next branch
  }

  // final store: out[r0 + v + 8h][ct*16 + c]
#pragma unroll
  for (int ct = 0; ct < 4; ++ct)
#pragma unroll
    for (int v = 0; v < 8; ++v)
      out[(size_t)(r0 + v + 8 * h) * DIM + ct * 16 + c] = oacc[ct][v];
}

// ---------------------------------------------------------------------------
// Launcher
// ---------------------------------------------------------------------------

extern "C" void kernel_launch(void* const* d_in, const int* in_sizes, int n_in,
                              void* d_out, int out_size, void* d_ws, size_t ws_size,
                              hipStream_t stream) {
  const float* x  = (const float*)d_in[0];
  const int*   ei = (const int*)d_in[1];  // [2][E]; harness delivers integer inputs as int32

  BranchParams p;  // setup order: hp, lp, i, hp2, lp2 at d_in[2 + 4j ...]
  for (int j = 0; j < 5; ++j) {
    p.W[j]  = (const float*)d_in[2 + 4 * j];
    p.b[j]  = (const float*)d_in[3 + 4 * j];
    p.wa[j] = (const float*)d_in[4 + 4 * j];
    p.ba[j] = (const float*)d_in[5 + 4 * j];
  }

  // workspace layout (floats): deg[N] | inv_deg[N] | ax[N*64] | aax[N*64]  (~26 MB)
  float* ws      = (float*)d_ws;
  float* deg     = ws;
  float* inv_deg = ws + N_NODES;
  float* ax      = ws + 2 * N_NODES;
  float* aax     = ax + (size_t)N_NODES * DIM;
  float* out     = (float*)d_out;

  hipMemsetAsync(deg, 0, (size_t)N_NODES * sizeof(float), stream);
  hipMemsetAsync(ax,  0, (size_t)N_NODES * DIM * sizeof(float), stream);
  hipMemsetAsync(aax, 0, (size_t)N_NODES * DIM * sizeof(float), stream);

  k_deg<<<(N_EDGES + 255) / 256, 256, 0, stream>>>(ei, deg);
  k_inv_deg<<<(N_NODES + 255) / 256, 256, 0, stream>>>(deg, inv_deg);

  const int scat_grid  = (N_EDGES * DIM + 255) / 256;
  const int scale_grid = (N_NODES * DIM + 255) / 256;

  k_scatter<<<scat_grid, 256, 0, stream>>>(x, ei, ax);    // A x
  k_scale<<<scale_grid, 256, 0, stream>>>(ax, inv_deg);   // ax = D^-1 A x
  k_scatter<<<scat_grid, 256, 0, stream>>>(ax, ei, aax);  // A ax
  k_scale<<<scale_grid, 256, 0, stream>>>(aax, inv_deg);  // aax = D^-1 A ax

  k_acm_wmma<<<N_NODES / 16, 32, 0, stream>>>(x, ax, aax, p, out);
}